// VAEBM_75703093559862
// MI455X (gfx1250) — compile-verified
//
#include <hip/hip_runtime.h>
#include <cstdint>

typedef float v2f __attribute__((ext_vector_type(2)));
typedef float v4f __attribute__((ext_vector_type(4)));
typedef float v8f __attribute__((ext_vector_type(8)));

#define HID   100
#define STEPS 100
#define KP    112   // padded K / row extent (7 tiles of 16, 28 chunks of 4)
#define NKC   28    // K chunks of 4 (f32 WMMA K=4)
#define NMT   7     // output-row tiles of 16
#define MST   114   // LDS row stride (floats) for M   (conflict-free mod 64)
#define WSTR  116   // per-point row stride (floats) for w staging (16B aligned, conflict-free)
#define SDE_WAVES 8
#define PTS_PER_WAVE 16

// fast sigmoid: v_exp_f32 + v_rcp_f32 (~1 ulp) instead of the 10-op IEEE divide
__device__ __forceinline__ float sigf(float x) {
  return __builtin_amdgcn_rcpf(1.0f + __expf(-x));
}

// ---------------------------------------------------------------------------
// Kernel 1: encoder  x(N,3) -> h(100) -> h(100) -> mu_zx(N,2), log_t_zx(N,1)
// 0.7 GFLOP total: VALU with weights in LDS is plenty; WMMA reserved for the
// 131-GFLOP SDE kernel below.  We2 is transposed into LDS so the inner
// reduction walks contiguous floats (merges into ds_load_b128).
// ---------------------------------------------------------------------------
__global__ __launch_bounds__(128) void vaebm_enc(
    const float* __restrict__ x,
    const float* __restrict__ We1, const float* __restrict__ be1,
    const float* __restrict__ We2, const float* __restrict__ be2,
    const float* __restrict__ Wmu, const float* __restrict__ bmu,
    const float* __restrict__ Wlt, const float* __restrict__ blt,
    float* __restrict__ out, int N)
{
  __shared__ float sWe1[3*HID], sbe1[HID], sWe2T[HID*HID], sbe2[HID], sWmu[2*HID], sWlt[HID];
  for (int i = threadIdx.x; i < HID*HID; i += blockDim.x) {
    int r = i / HID, c = i - r*HID;          // sWe2T[j2][j1] = We2[j1][j2]
    sWe2T[i] = We2[c*HID + r];
  }
  for (int i = threadIdx.x; i < 3*HID;   i += blockDim.x) sWe1[i] = We1[i];
  for (int i = threadIdx.x; i < HID;     i += blockDim.x) {
    sbe1[i] = be1[i]; sbe2[i] = be2[i]; sWlt[i] = Wlt[i];
    sWmu[2*i] = Wmu[2*i]; sWmu[2*i+1] = Wmu[2*i+1];
  }
  __syncthreads();
  int pt = blockIdx.x * blockDim.x + threadIdx.x;
  if (pt >= N) return;
  float x0 = x[3*pt], x1 = x[3*pt+1], x2 = x[3*pt+2];
  float h1[HID];
#pragma unroll
  for (int j = 0; j < HID; ++j) {
    float a = fmaf(x0, sWe1[j], fmaf(x1, sWe1[HID+j], fmaf(x2, sWe1[2*HID+j], sbe1[j])));
    h1[j] = a * sigf(a);
  }
  float mu0 = bmu[0], mu1 = bmu[1], lt = blt[0];
#pragma unroll 1
  for (int j2 = 0; j2 < HID; ++j2) {
    float a = sbe2[j2];
#pragma unroll
    for (int j1 = 0; j1 < HID; ++j1) a = fmaf(h1[j1], sWe2T[j2*HID + j1], a);
    float h2 = a * sigf(a);
    mu0 = fmaf(h2, sWmu[2*j2],   mu0);
    mu1 = fmaf(h2, sWmu[2*j2+1], mu1);
    lt  = fmaf(h2, sWlt[j2],     lt);
  }
  out[(size_t)8*N + 2*pt]   = mu0;   // mu_zx
  out[(size_t)8*N + 2*pt+1] = mu1;
  out[(size_t)10*N + pt]    = lt;    // log_t_zx
}

// ---------------------------------------------------------------------------
// Kernel 2: 100-step Riemannian Brownian motion.
// Math restructuring: G = u^T M u, dG = v^T M u + u^T M v with fixed symmetric
// M = Wdmu Wdmu^T + Wds Wds^T (100x100).  Per step the only big contraction is
// w_m = M u_m, done as a shared-A fp32 GEMM with V_WMMA_F32_16X16X4_F32:
// each wave = 16 points, lane halves carry K%4 halves exactly as the WMMA
// A/B f32 layouts demand (lanes 0-15: K={k,k+1}; lanes 16-31: K={k+2,k+3}).
// Wpk fetches are explicit 16B vector loads -> single ds_load_b128 each.
// ---------------------------------------------------------------------------
__global__ __launch_bounds__(SDE_WAVES*32) void vaebm_sde(
    const float* __restrict__ Wd1, const float* __restrict__ bd1,
    const float* __restrict__ Wdmu, const float* __restrict__ Wds,
    const float* __restrict__ noise,
    float* __restrict__ out, int N)
{
  extern __shared__ float smem[];
  float* Mlds = smem;                    // KP * MST     (zero-padded, symmetric)
  float* Wpk  = Mlds + KP*MST;           // KP * 4: {Wd1[0,j], Wd1[1,j], bd1[j], 0}
  float* wst  = Wpk + KP*4;              // SDE_WAVES * 2 * 16 * WSTR  w staging

  // Build M once per block (trivial: 12768 entries * 6 FLOP).
  for (int i = threadIdx.x; i < KP*MST; i += blockDim.x) {
    int r = i / MST, c = i - r*MST;
    float v = 0.0f;
    if (r < HID && c < HID) {
#pragma unroll
      for (int q = 0; q < 3; ++q)
        v += Wdmu[3*r+q]*Wdmu[3*c+q] + Wds[3*r+q]*Wds[3*c+q];
    }
    Mlds[i] = v;
  }
  for (int j = threadIdx.x; j < KP; j += blockDim.x) {
    float w0 = 0.0f, w1 = 0.0f, bb = 0.0f;
    if (j < HID) { w0 = Wd1[j]; w1 = Wd1[HID+j]; bb = bd1[j]; }
    Wpk[4*j] = w0; Wpk[4*j+1] = w1; Wpk[4*j+2] = bb; Wpk[4*j+3] = 0.0f;
  }
  __syncthreads();

  const int lane = threadIdx.x & 31;
  const int wave = threadIdx.x >> 5;
  const int lh   = lane >> 4;            // lane half: selects K pair within chunk
  const int p    = lane & 15;            // point column within wave
  const int pt   = blockIdx.x*(SDE_WAVES*PTS_PER_WAVE) + wave*PTS_PER_WAVE + p;
  const int jhalf = 2*lh;

  float* ww = wst + wave*(2*16*WSTR);    // per-wave staging [n][p][row]

  float z0  = out[(size_t)8*N + 2*pt];   // z init = mu_zx (written by enc)
  float z1  = out[(size_t)8*N + 2*pt+1];
  float ltv = out[(size_t)10*N + pt];
  float sqdt = sqrtf(__expf(2.0f*ltv) * (1.0f/STEPS));

#pragma unroll 1
  for (int s = 0; s < STEPS; ++s) {
    // ---- pass 1: B-operands u0,u1 in registers (branchless; pads are 0 via Wpk) ----
    float u0[2*NKC], u1[2*NKC];
#pragma unroll
    for (int li = 0; li < 2*NKC; ++li) {
      int kc = li >> 1, t = li & 1;
      int j = 4*kc + jhalf + t;                       // <= 111 = KP-1
      v4f wp = *(const v4f*)(Wpk + 4*j);              // one ds_load_b128
      float a = fmaf(z0, wp[0], fmaf(z1, wp[1], wp[2]));
      float e = sigf(a);
      float s1 = e * fmaf(a, 1.0f - e, 1.0f);         // swish'
      u0[li] = wp[0] * s1;                             // 0 for j>=100 (wp==0)
      u1[li] = wp[1] * s1;
    }

    // ---- pass 2: w_n = M @ u_n via fp32 WMMA (7 tiles x 28 K-chunks x 2 cols) ----
    const float* Abase = Mlds + (unsigned)(lane & 15)*MST + jhalf;
#pragma unroll
    for (int mt = 0; mt < NMT; ++mt) {
      v8f acc0 = {}; v8f acc1 = {};
      const float* Ap = Abase + mt*16*MST;
#pragma unroll
      for (int kc = 0; kc < NKC; ++kc) {
        v2f A = *(const v2f*)(Ap + 4*kc);             // A[m_row, K pair]
        v2f B0; B0[0] = u0[2*kc]; B0[1] = u0[2*kc+1]; // B[K pair, point p]
        v2f B1; B1[0] = u1[2*kc]; B1[1] = u1[2*kc+1];
        acc0 = __builtin_amdgcn_wmma_f32_16x16x4_f32(false, A, false, B0,
                                                     (short)0, acc0, false, false);
        acc1 = __builtin_amdgcn_wmma_f32_16x16x4_f32(false, A, false, B1,
                                                     (short)0, acc1, false, false);
      }
      // C layout: lane(lh,p) holds rows mt*16 + 8*lh + (0..7), column p
      float* wb0 = ww + p*WSTR + mt*16 + 8*lh;
      float* wb1 = wb0 + 16*WSTR;
      v4f t0, t1;
      t0[0]=acc0[0]; t0[1]=acc0[1]; t0[2]=acc0[2]; t0[3]=acc0[3];
      t1[0]=acc0[4]; t1[1]=acc0[5]; t1[2]=acc0[6]; t1[3]=acc0[7];
      *(v4f*)wb0 = t0; *(v4f*)(wb0+4) = t1;
      t0[0]=acc1[0]; t0[1]=acc1[1]; t0[2]=acc1[2]; t0[3]=acc1[3];
      t1[0]=acc1[4]; t1[1]=acc1[5]; t1[2]=acc1[6]; t1[3]=acc1[7];
      *(v4f*)wb1 = t0; *(v4f*)(wb1+4) = t1;
    }

    // ---- pass 3: the 9 length-100 dots against w, then 2x2 geometry ----
    float g00=0,g01=0,g11=0;
    float q00_0=0,q00_1=0,q01_0=0,q01_1=0,q11_0=0,q11_1=0;
    const float* w0base = ww + p*WSTR + jhalf;
    const float* w1base = w0base + 16*WSTR;
#pragma unroll
    for (int kc = 0; kc < NKC; ++kc) {
      v2f w0p = *(const v2f*)(w0base + 4*kc);
      v2f w1p = *(const v2f*)(w1base + 4*kc);
#pragma unroll
      for (int t = 0; t < 2; ++t) {
        int j = 4*kc + jhalf + t;
        v4f wp = *(const v4f*)(Wpk + 4*j);            // one ds_load_b128
        float a = fmaf(z0, wp[0], fmaf(z1, wp[1], wp[2]));
        float e = sigf(a);
        float s1 = e * fmaf(a, 1.0f - e, 1.0f);                   // swish'
        float s2 = e * (1.0f - e) * fmaf(a, fmaf(-2.0f, e, 1.0f), 2.0f); // swish''
        float uu0 = wp[0]*s1, uu1 = wp[1]*s1;
        float v00 = wp[0]*wp[0]*s2, v01 = wp[0]*wp[1]*s2, v11 = wp[1]*wp[1]*s2;
        float wa = w0p[t], wb = w1p[t];
        g00 = fmaf(uu0, wa, g00); g01 = fmaf(uu0, wb, g01); g11 = fmaf(uu1, wb, g11);
        q00_0 = fmaf(v00, wa, q00_0); q00_1 = fmaf(v00, wb, q00_1);
        q01_0 = fmaf(v01, wa, q01_0); q01_1 = fmaf(v01, wb, q01_1);
        q11_0 = fmaf(v11, wa, q11_0); q11_1 = fmaf(v11, wb, q11_1);
      }
    }
    // lane pair (L, L^16) holds complementary j-subsets: pair-reduce
    g00  += __shfl_xor(g00, 16, 32);  g01  += __shfl_xor(g01, 16, 32);
    g11  += __shfl_xor(g11, 16, 32);
    q00_0 += __shfl_xor(q00_0,16,32); q00_1 += __shfl_xor(q00_1,16,32);
    q01_0 += __shfl_xor(q01_0,16,32); q01_1 += __shfl_xor(q01_1,16,32);
    q11_0 += __shfl_xor(q11_0,16,32); q11_1 += __shfl_xor(q11_1,16,32);

    float detg = fmaf(g00, g11, -g01*g01);
    float inv  = __builtin_amdgcn_rcpf(detg);         // v_rcp_f32, not IEEE div
    float gi[2][2];
    gi[0][0] = g11*inv; gi[0][1] = -g01*inv; gi[1][0] = gi[0][1]; gi[1][1] = g00*inv;

    float D[2][2][2];                       // D[k][m][l] = dG[k,m]/dz_l
    D[0][0][0] = 2.0f*q00_0;    D[0][0][1] = 2.0f*q01_0;
    D[0][1][0] = q00_1 + q01_0; D[0][1][1] = q01_1 + q11_0;
    D[1][0][0] = D[0][1][0];    D[1][0][1] = D[0][1][1];
    D[1][1][0] = 2.0f*q01_1;    D[1][1][1] = 2.0f*q11_1;

    float dr[2];
#pragma unroll
    for (int i = 0; i < 2; ++i) {
      float di = 0.0f;
#pragma unroll
      for (int k = 0; k < 2; ++k)
#pragma unroll
        for (int l = 0; l < 2; ++l) {
          float ch = 0.0f;
#pragma unroll
          for (int m = 0; m < 2; ++m)
            ch += gi[i][m] * (D[k][m][l] + D[l][m][k] - D[k][l][m]);
          di += gi[k][l] * (0.5f * ch);     // Chris[i][k][l]
        }
      dr[i] = 0.5f * di;
    }

    v2f nz = *(const v2f*)(noise + ((size_t)s*N + pt)*2);
    float dW0 = sqdt*nz[0], dW1 = sqdt*nz[1];
    z0 += dr[0] + gi[0][0]*dW0 + gi[0][1]*dW1;
    z1 += dr[1] + gi[1][0]*dW0 + gi[1][1]*dW1;

    int sn = (s+1 < STEPS) ? (s+1) : s;     // uniform; prefetch next step's noise
    __builtin_prefetch(noise + ((size_t)sn*N + pt)*2, 0, 1);
  }
  if (lh == 0) {                            // one lane per point writes z
    out[2*pt]   = z0;
    out[2*pt+1] = z1;
  }
}

// ---------------------------------------------------------------------------
// Kernel 3: final decoder heads on z, plus prior broadcasts.
// ---------------------------------------------------------------------------
__global__ __launch_bounds__(256) void vaebm_dec(
    const float* __restrict__ Wd1, const float* __restrict__ bd1,
    const float* __restrict__ Wdmu, const float* __restrict__ bdmu,
    const float* __restrict__ Wds, const float* __restrict__ bds,
    const float* __restrict__ bmuz, const float* __restrict__ bltz,
    float* __restrict__ out, int N)
{
  __shared__ float sWd1[2*HID], sbd1[HID], sWdmu[3*HID], sWds[3*HID];
  for (int i = threadIdx.x; i < 2*HID; i += blockDim.x) sWd1[i] = Wd1[i];
  for (int i = threadIdx.x; i < HID;   i += blockDim.x) sbd1[i] = bd1[i];
  for (int i = threadIdx.x; i < 3*HID; i += blockDim.x) { sWdmu[i] = Wdmu[i]; sWds[i] = Wds[i]; }
  __syncthreads();
  int pt = blockIdx.x*blockDim.x + threadIdx.x;
  if (pt >= N) return;
  float z0 = out[2*pt], z1 = out[2*pt+1];
  float m0 = bdmu[0], m1 = bdmu[1], m2 = bdmu[2];
  float l0 = bds[0],  l1 = bds[1],  l2 = bds[2];
#pragma unroll 4
  for (int j = 0; j < HID; ++j) {
    float a = fmaf(z0, sWd1[j], fmaf(z1, sWd1[HID+j], sbd1[j]));
    float h = a * sigf(a);
    m0 = fmaf(h, sWdmu[3*j],   m0);
    m1 = fmaf(h, sWdmu[3*j+1], m1);
    m2 = fmaf(h, sWdmu[3*j+2], m2);
    l0 = fmaf(h, sWds[3*j],    l0);
    l1 = fmaf(h, sWds[3*j+1],  l1);
    l2 = fmaf(h, sWds[3*j+2],  l2);
  }
  out[(size_t)2*N + 3*pt]   = m0;   // mu_xz
  out[(size_t)2*N + 3*pt+1] = m1;
  out[(size_t)2*N + 3*pt+2] = m2;
  out[(size_t)5*N + 3*pt]   = l0;   // log_sigma_xz
  out[(size_t)5*N + 3*pt+1] = l1;
  out[(size_t)5*N + 3*pt+2] = l2;
  out[(size_t)11*N + 2*pt]   = bmuz[0];  // mu_z
  out[(size_t)11*N + 2*pt+1] = bmuz[1];
  out[(size_t)13*N + pt]     = bltz[0];  // log_t_z
}

extern "C" void kernel_launch(void* const* d_in, const int* in_sizes, int n_in,
                              void* d_out, int out_size, void* d_ws, size_t ws_size,
                              hipStream_t stream)
{
  const float* x    = (const float*)d_in[0];
  const float* We1  = (const float*)d_in[1];
  const float* be1  = (const float*)d_in[2];
  const float* We2  = (const float*)d_in[3];
  const float* be2  = (const float*)d_in[4];
  const float* Wmu  = (const float*)d_in[5];
  const float* bmu  = (const float*)d_in[6];
  const float* Wlt  = (const float*)d_in[7];
  const float* blt  = (const float*)d_in[8];
  const float* Wd1  = (const float*)d_in[9];
  const float* bd1  = (const float*)d_in[10];
  const float* Wdmu = (const float*)d_in[11];
  const float* bdmu = (const float*)d_in[12];
  const float* Wds  = (const float*)d_in[13];
  const float* bds  = (const float*)d_in[14];
  const float* bmuz = (const float*)d_in[15];
  const float* bltz = (const float*)d_in[16];
  const float* noise= (const float*)d_in[17];
  float* out = (float*)d_out;
  const int N = in_sizes[0] / 3;   // 32768

  const size_t SMEM = (size_t)(KP*MST + KP*4 + SDE_WAVES*2*16*WSTR) * sizeof(float);
  (void)hipFuncSetAttribute(reinterpret_cast<const void*>(vaebm_sde),
                            hipFuncAttributeMaxDynamicSharedMemorySize, (int)SMEM);

  vaebm_enc<<<N/128, 128, 0, stream>>>(x, We1, be1, We2, be2, Wmu, bmu, Wlt, blt, out, N);
  vaebm_sde<<<N/(SDE_WAVES*PTS_PER_WAVE), SDE_WAVES*32, SMEM, stream>>>(
      Wd1, bd1, Wdmu, Wds, noise, out, N);
  vaebm_dec<<<N/256, 256, 0, stream>>>(Wd1, bd1, Wdmu, bdmu, Wds, bds, bmuz, bltz, out, N);
}